// SequenceFocalLoss_79422535238404
// MI455X (gfx1250) — compile-verified
//
#include <hip/hip_runtime.h>
#include <stdint.h>

#define B_  16
#define N_  100000
#define C_  10
#define M_  32

typedef float v2f  __attribute__((ext_vector_type(2)));
typedef float v4f  __attribute__((ext_vector_type(4)));
typedef float v8f  __attribute__((ext_vector_type(8)));
typedef unsigned int u32x4 __attribute__((ext_vector_type(4)));
typedef int   i32x4 __attribute__((ext_vector_type(4)));
typedef int   i32x8 __attribute__((ext_vector_type(8)));

#define IPT 4              // anchors per thread
#define TPB 256            // threads per block (8 wave32)
#define STRIDE (98 * TPB)  // 98 blocks/batch * 256 = 25088

// ---------------------------------------------------------------- init
__global__ void fl_init(float* __restrict__ acc) {
  int t = threadIdx.x;
  if (t < 3 * B_) acc[t] = 0.0f;
}

// ---------------------------------------------------------------- main
__global__ __launch_bounds__(TPB) void fl_main(
    const float* __restrict__ cls, const float* __restrict__ reg,
    const float* __restrict__ anc, const float* __restrict__ ann,
    float* __restrict__ acc) {
  const int b = blockIdx.y;
  const int t = threadIdx.x;

  __shared__ __align__(16) float s_raw[M_ * 5];   // 640 B, staged by TDM + async
  __shared__ __align__(16) v4f   s_box[M_];       // {x1,y1,x2,y2} (poisoned if invalid)
  __shared__ __align__(8)  v2f   s_al[M_];        // {area, label}
  __shared__ float s_red[3 * 8];

  // ---- CDNA5 TDM: descriptor-driven DMA of the annotation tile to LDS ----
#if __has_builtin(__builtin_amdgcn_tensor_load_to_lds)
  if (t == 0) {  // one wave issues; TDM ignores EXEC
    uint64_t ga = (uint64_t)(uintptr_t)(ann + (size_t)b * M_ * 5);
    uint32_t la = (uint32_t)(uintptr_t)(&s_raw[0]);
    u32x4 g0;
    g0[0] = 1u;                                        // count=1 (valid user D#)
    g0[1] = la;                                        // lds_addr
    g0[2] = (uint32_t)(ga & 0xFFFFFFFFu);              // global_addr[31:0]
    g0[3] = (uint32_t)((ga >> 32) & 0x1FFFFFFu)        // global_addr[56:32]
          | (2u << 30);                                // type=2 ("image")
    i32x8 g1;
    g1[0] = (int)(2u << 16);                           // data_size=4B; mask=0
    g1[1] = (int)((160u & 0xFFFFu) << 16);             // tensor_dim0=160 (lo16)
    g1[2] = (int)(1u << 16);                           // dim0 hi=0; tensor_dim1=1 (lo16)
    g1[3] = (int)(160u << 16);                         // dim1 hi=0; tile_dim0=160
    g1[4] = 0;                                         // tile_dim1=0, tile_dim2=0
    g1[5] = (int)160u;                                 // tensor_dim0_stride lo32
    g1[6] = 0;                                         // stride0 hi; stride1 lo=0
    g1[7] = 0;
    i32x4 z4; z4[0] = 0; z4[1] = 0; z4[2] = 0; z4[3] = 0;
    i32x8 z8; z8[0] = 0; z8[1] = 0; z8[2] = 0; z8[3] = 0;
    z8[4] = 0; z8[5] = 0; z8[6] = 0; z8[7] = 0;
    // 6-arg toolchain form: (g0, g1, g2, g3, g4, cpol)
    __builtin_amdgcn_tensor_load_to_lds(g0, g1, z4, z4, z8, 0);
  }
#endif

  // ---- CDNA5 async global->LDS copy (40 x 16B), same bytes (benign dup) ----
  if (t < 40) {
    uint32_t loff = (uint32_t)(uintptr_t)(&s_raw[0]) + (uint32_t)(t * 16);
    const void* g = (const void*)((const char*)(ann + (size_t)b * M_ * 5) + t * 16);
    asm volatile("global_load_async_to_lds_b128 %0, %1, off"
                 :: "v"(loff), "v"(g) : "memory");
  }
  asm volatile("s_wait_asynccnt 0" ::: "memory");
#if __has_builtin(__builtin_amdgcn_s_wait_tensorcnt)
  __builtin_amdgcn_s_wait_tensorcnt(0);
#else
  asm volatile("s_wait_tensorcnt 0" ::: "memory");
#endif
  __syncthreads();

  if (t < M_) {
    float x1 = s_raw[t * 5 + 0], y1 = s_raw[t * 5 + 1];
    float x2 = s_raw[t * 5 + 2], y2 = s_raw[t * 5 + 3];
    float lb = s_raw[t * 5 + 4];
    bool valid = (lb != -1.0f);
    v4f bx; v2f al;
    if (valid) {
      bx.x = x1; bx.y = y1; bx.z = x2; bx.w = y2;
      al.x = (x2 - x1) * (y2 - y1); al.y = lb;
    } else {
      // poisoned box: forces inter=0, union~1e30 (iou~0); matches ref masking
      bx.x = 1e30f; bx.y = 1e30f; bx.z = -1e30f; bx.w = -1e30f;
      al.x = 1e30f; al.y = 0.0f;
    }
    s_box[t] = bx; s_al[t] = al;
  }
  __syncthreads();

  float c_cls = 0.0f, c_reg = 0.0f, c_pos = 0.0f;

  const int base = blockIdx.x * TPB + t;          // [0, 25088)
  int  idx[IPT];
  bool ok[IPT];
  v4f  a[IPT];
  float aw[IPT], ah[IPT], areaA[IPT];
  float bi[IPT], bu[IPT];
  int   bidx[IPT];

  #pragma unroll
  for (int k = 0; k < IPT; ++k) {
    idx[k] = base + k * STRIDE;
    ok[k]  = (idx[k] < N_);
    if (ok[k]) {
      a[k] = __builtin_nontemporal_load((const v4f*)anc + ((size_t)b * N_ + idx[k]));
    } else {
      a[k].x = 0.f; a[k].y = 0.f; a[k].z = 0.f; a[k].w = 0.f;
    }
    aw[k] = a[k].z - a[k].x;
    ah[k] = a[k].w - a[k].y;
    areaA[k] = aw[k] * ah[k];
    bi[k] = -1.0f; bu[k] = 1.0f; bidx[k] = 0;
  }

  // ---- IoU argmax over 32 GT boxes, division-free cross-mult compare ----
  #pragma unroll 4
  for (int j = 0; j < M_; ++j) {
    v4f bx = s_box[j];           // ds_load_b128, broadcast
    v2f al = s_al[j];            // ds_load_b64, broadcast
    #pragma unroll
    for (int k = 0; k < IPT; ++k) {
      float iw = fminf(a[k].z, bx.z) - fmaxf(a[k].x, bx.x);
      float ih = fminf(a[k].w, bx.w) - fmaxf(a[k].y, bx.y);
      float inter = fmaxf(iw, 0.0f) * fmaxf(ih, 0.0f);
      float un = areaA[k] + al.x - inter;     // > 0 always (boxes non-degenerate)
      bool better = inter * bu[k] > bi[k] * un;
      bi[k]   = better ? inter : bi[k];
      bu[k]   = better ? un    : bu[k];
      bidx[k] = better ? j     : bidx[k];
    }
  }

  // ---- per-anchor losses ----
  #pragma unroll
  for (int k = 0; k < IPT; ++k) {
    if (!ok[k]) continue;
    float iou = bi[k] / fmaxf(bu[k], 1e-8f);
    bool pos = iou >= 0.5f;
    bool bg  = iou < 0.4f;
    const size_t ai = (size_t)b * N_ + idx[k];

    if (pos || bg) {
      int lab = pos ? (int)s_al[bidx[k]].y : -1;
      const v2f* cp = (const v2f*)(cls + ai * C_);    // 8B-aligned rows
      float s = 0.0f;
      #pragma unroll
      for (int h = 0; h < 5; ++h) {
        v2f pv = __builtin_nontemporal_load(cp + h);
        #pragma unroll
        for (int q = 0; q < 2; ++q) {
          int c = 2 * h + q;
          float p = fminf(fmaxf(pv[q], 1e-4f), 1.0f - 1e-4f);
          if (c == lab) s += 0.25f * (1.0f - p) * (1.0f - p) * (-__logf(p));
          else          s += 0.75f * p * p * (-__logf(1.0f - p));
        }
      }
      c_cls += s;
    }

    if (pos) {
      c_pos += 1.0f;
      v4f r = __builtin_nontemporal_load((const v4f*)reg + ai);
      v4f g = s_box[bidx[k]];
      float acx = a[k].x + 0.5f * aw[k], acy = a[k].y + 0.5f * ah[k];
      float gw = g.z - g.x, gh = g.w - g.y;
      float gcx = g.x + 0.5f * gw, gcy = g.y + 0.5f * gh;
      gw = fmaxf(gw, 1.0f); gh = fmaxf(gh, 1.0f);
      float t0 = ((gcx - acx) / aw[k]) / 0.1f;
      float t1 = ((gcy - acy) / ah[k]) / 0.1f;
      float t2 = __logf(gw / aw[k]) / 0.2f;
      float t3 = __logf(gh / ah[k]) / 0.2f;
      float d0 = fabsf(t0 - r.x), d1 = fabsf(t1 - r.y);
      float d2 = fabsf(t2 - r.z), d3 = fabsf(t3 - r.w);
      const float th = 1.0f / 9.0f, hc = 0.5f / 9.0f;
      c_reg += (d0 < th ? 4.5f * d0 * d0 : d0 - hc);
      c_reg += (d1 < th ? 4.5f * d1 * d1 : d1 - hc);
      c_reg += (d2 < th ? 4.5f * d2 * d2 : d2 - hc);
      c_reg += (d3 < th ? 4.5f * d3 * d3 : d3 - hc);
    }
  }

  // ---- intra-wave reduction (wave32 shuffles) ----
  float v0 = c_cls, v1 = c_reg, v2 = c_pos;
  #pragma unroll
  for (int o = 16; o > 0; o >>= 1) {
    v0 += __shfl_xor(v0, o, 32);
    v1 += __shfl_xor(v1, o, 32);
    v2 += __shfl_xor(v2, o, 32);
  }
  int wave = t >> 5, lane = t & 31;
  if (lane == 0) { s_red[wave] = v0; s_red[8 + wave] = v1; s_red[16 + wave] = v2; }
  __syncthreads();

  // ---- cross-wave reduction: one V_WMMA_F32_16X16X4_F32 on wave 0 ----
  // A[m,k] (m = q*2+r, q=quantity, r=half): partials P[q][4r+k]; B = ones;
  // D[m,n] = sum_k A[m,k].  Lane 0 holds D[0..7, 0] in its 8 result VGPRs.
#if __has_builtin(__builtin_amdgcn_wmma_f32_16x16x4_f32)
  if (t < 32) {   // full wave 0: EXEC all ones (WMMA requirement)
    int m  = t & 15;
    int kb = (t >> 4) * 2;          // lanes 0-15: K=0,1; lanes 16-31: K=2,3
    int q  = m >> 1, r = m & 1;
    v2f A; A.x = 0.0f; A.y = 0.0f;
    if (m < 6) {
      A.x = s_red[q * 8 + 4 * r + kb];
      A.y = s_red[q * 8 + 4 * r + kb + 1];
    }
    v2f Bv; Bv.x = 1.0f; Bv.y = 1.0f;   // all-ones B: layout independent
    v8f Cv = {};
    v8f D = __builtin_amdgcn_wmma_f32_16x16x4_f32(
        false, A, false, Bv, (short)0, Cv, false, false);
    if (t == 0) {
      atomicAdd(&acc[b],          D[0] + D[1]);
      atomicAdd(&acc[B_ + b],     D[2] + D[3]);
      atomicAdd(&acc[2 * B_ + b], D[4] + D[5]);
    }
  }
#else
  if (t == 0) {
    float a0 = 0.f, a1 = 0.f, a2 = 0.f;
    #pragma unroll
    for (int w = 0; w < TPB / 32; ++w) {
      a0 += s_red[w]; a1 += s_red[8 + w]; a2 += s_red[16 + w];
    }
    atomicAdd(&acc[b], a0);
    atomicAdd(&acc[B_ + b], a1);
    atomicAdd(&acc[2 * B_ + b], a2);
  }
#endif
}

// ---------------------------------------------------------------- finalize
__global__ void fl_final(const float* __restrict__ ann,
                         const float* __restrict__ acc,
                         float* __restrict__ out) {
  int b = threadIdx.x;  // 32 lanes, 16 used
  float clsl = 0.0f, regl = 0.0f, has = 0.0f;
  if (b < B_) {
    int anyv = 0;
    for (int m = 0; m < M_; ++m)
      anyv |= (ann[(size_t)b * M_ * 5 + m * 5 + 4] != -1.0f) ? 1 : 0;
    has = (float)anyv;
    float np = acc[2 * B_ + b];
    if (np > 0.0f) {
      float npf = fmaxf(np, 1.0f);
      clsl = acc[b] / npf;
      regl = acc[B_ + b] / fmaxf(npf * 4.0f, 1.0f);
    }
  }
  float sc = clsl * has, sr = regl * has, sh = has;
  #pragma unroll
  for (int o = 16; o > 0; o >>= 1) {
    sc += __shfl_xor(sc, o, 32);
    sr += __shfl_xor(sr, o, 32);
    sh += __shfl_xor(sh, o, 32);
  }
  if (b == 0) {
    float den = fmaxf(sh, 1.0f);
    out[0] = sc / den;
    out[1] = sr / den;
  }
}

// ---------------------------------------------------------------- launch
extern "C" void kernel_launch(void* const* d_in, const int* in_sizes, int n_in,
                              void* d_out, int out_size, void* d_ws, size_t ws_size,
                              hipStream_t stream) {
  (void)in_sizes; (void)n_in; (void)out_size; (void)ws_size;
  const float* cls = (const float*)d_in[0];
  const float* reg = (const float*)d_in[1];
  const float* anc = (const float*)d_in[2];
  const float* ann = (const float*)d_in[3];
  float* out = (float*)d_out;
  float* acc = (float*)d_ws;   // [3*B_] accumulators

  fl_init<<<1, 64, 0, stream>>>(acc);
  dim3 grid(98, B_);           // 98*256*4 >= 100000 anchors per batch
  fl_main<<<grid, TPB, 0, stream>>>(cls, reg, anc, ann, acc);
  fl_final<<<1, 32, 0, stream>>>(ann, acc, out);
}